// SpatialTransformer_19164144075056
// MI455X (gfx1250) — compile-verified
//
#include <hip/hip_runtime.h>

// SpatialTransformer 3D trilinear warp, border padding, align_corners=True.
// src:  [B=2, C=1, D=160, H=192, W=224] f32
// flow: [B=2, 3,   D,     H,     W]     f32  (channels: d, h, w displacement)
// out:  [B=2, C=1, D,     H,     W]     f32
//
// Memory-bound. HBM floor ~275MB -> ~12us @ 23.3 TB/s, but the practical
// limiter is gather traffic L2->WGP (one ~64B line per 8B tap). Mitigations:
// - src (55MB) is L2-resident (192MB L2); flow/out streamed with TH_NT so the
//   streams never evict it.
// - Wave-compact 3D tiling: blockDim=(8,8,4), each wave covers a 32x4x1
//   (WxHxD) voxel patch -> per-instruction lane coalescing merges gather
//   lines (~2x less L2 sector traffic than a 128-wide row wave).
// - 4 voxels/thread along W: b128 flow loads / b128 store; the two W-adjacent
//   taps fused into one b64 pair load (4 gathers/voxel).
// - VALU-lean math: trunc==floor (coords >= 0), v_fract pattern for
//   fractions, mask&AND corner deltas, byte offsets straight into voffset.

#define Bc 2
#define Dc 160
#define Hc 192
#define Wc 224

__device__ __host__ constexpr int cHW    = Hc * Wc;      // 43008
__device__ __host__ constexpr int cDHW   = Dc * cHW;     // 6881280
__device__ __host__ constexpr int cFLOWB = 3 * cDHW;     // per-batch flow stride
__device__ __host__ constexpr int cHWb   = cHW * 4;      // row-plane stride, bytes
__device__ __host__ constexpr int cWb    = Wc * 4;       // row stride, bytes

typedef float f4 __attribute__((ext_vector_type(4)));
typedef float f2 __attribute__((ext_vector_type(2)));
typedef int   i4 __attribute__((ext_vector_type(4)));

// Two consecutive floats at (base + byte offset), 4B-aligned (one b64 gather).
__device__ __forceinline__ f2 ldpair_b(const float* __restrict__ base, int byteoff) {
    f2 r;
    __builtin_memcpy(&r, (const char*)base + byteoff, sizeof(f2));
    return r;
}

__global__ __launch_bounds__(256) void spatial_transformer_warp(
        const float* __restrict__ src,
        const float* __restrict__ flow,
        float* __restrict__ out) {
    // blockDim=(8,8,4): block tile = 32(W) x 8(H) x 4(D) voxels.
    // grid=(W/32, H/8, (D/4)*B) = (7, 24, 80).
    const int w  = ((int)blockIdx.x * 8 + (int)threadIdx.x) * 4;   // 0..220
    const int h  = (int)blockIdx.y * 8 + (int)threadIdx.y;         // 0..191
    const int zb = (int)blockIdx.z;                                // 0..79
    const int b  = (zb >= Dc / 4) ? 1 : 0;                         // scalar
    const int d  = (zb - b * (Dc / 4)) * 4 + (int)threadIdx.z;     // 0..159

    const int r = d * cHW + h * Wc + w;                            // elem index [D,H,W]

    // Streaming flow reads (used once): non-temporal, keep L2 for src.
    const float* fb = flow + (size_t)b * cFLOWB + r;
    f4 dd4 = __builtin_nontemporal_load((const f4*)(fb));
    f4 dh4 = __builtin_nontemporal_load((const f4*)(fb + cDHW));
    f4 dw4 = __builtin_nontemporal_load((const f4*)(fb + 2 * cDHW));

    // Clamped sample coordinates (SoA over the 4 voxels). min(max()) -> v_med3.
    const f4 zero = 0.0f;
    f4 wv  = {(float)w, (float)(w + 1), (float)(w + 2), (float)(w + 3)};
    f4 cd4 = __builtin_elementwise_min(
                 __builtin_elementwise_max(dd4 + (f4)((float)d), zero),
                 (f4)((float)(Dc - 1)));
    f4 ch4 = __builtin_elementwise_min(
                 __builtin_elementwise_max(dh4 + (f4)((float)h), zero),
                 (f4)((float)(Hc - 1)));
    f4 cw4 = __builtin_elementwise_min(
                 __builtin_elementwise_max(dw4 + wv, zero),
                 (f4)((float)(Wc - 1)));

    // Coordinates are >= 0, so fptosi truncation == floor.
    i4 d0 = __builtin_convertvector(cd4, i4);
    i4 h0 = __builtin_convertvector(ch4, i4);
    i4 w0 = __builtin_convertvector(cw4, i4);

    // Fractions: x - floor(x) folds to v_fract_f32.
    f4 fd = cd4 - __builtin_elementwise_floor(cd4);
    f4 fh = ch4 - __builtin_elementwise_floor(ch4);

    // W pair base clamped to W-2; fraction re-based on wp so the border case
    // (w0 == W-1 -> cw == W-1 exactly) gives fw == 1.0: x-lerp lands on the
    // last element with no conditional selects.
    i4 wp = __builtin_elementwise_min(w0, (i4)(Wc - 2));
    f4 fw = cw4 - __builtin_convertvector(wp, f4);

    // Corner byte offsets: base once; +row / +plane deltas via compare-mask.
    i4 b00 = d0 * cHWb + h0 * cWb + (wp << 2);
    i4 hm  = h0 < (i4)(Hc - 1);          // -1 / 0 lane mask
    i4 dm  = d0 < (i4)(Dc - 1);
    i4 dhs = hm & (i4)cWb;               // 0 or W*4
    i4 dds = dm & (i4)cHWb;              // 0 or HW*4
    i4 b01 = b00 + dhs;
    i4 b10 = b00 + dds;
    i4 b11 = b01 + dds;

    const float* sb = src + (size_t)b * cDHW;                      // C == 1

    // 16 b64 pair gathers (L2-resident src).
    f2 p00[4], p01[4], p10[4], p11[4];
#pragma unroll
    for (int j = 0; j < 4; ++j) {
        p00[j] = ldpair_b(sb, b00[j]);
        p01[j] = ldpair_b(sb, b01[j]);
        p10[j] = ldpair_b(sb, b10[j]);
        p11[j] = ldpair_b(sb, b11[j]);
    }

    // Re-pack to SoA; 7 lerps in f4 (packed / VOPD dual-issue friendly).
    f4 V000 = {p00[0].x, p00[1].x, p00[2].x, p00[3].x};
    f4 V001 = {p00[0].y, p00[1].y, p00[2].y, p00[3].y};
    f4 V010 = {p01[0].x, p01[1].x, p01[2].x, p01[3].x};
    f4 V011 = {p01[0].y, p01[1].y, p01[2].y, p01[3].y};
    f4 V100 = {p10[0].x, p10[1].x, p10[2].x, p10[3].x};
    f4 V101 = {p10[0].y, p10[1].y, p10[2].y, p10[3].y};
    f4 V110 = {p11[0].x, p11[1].x, p11[2].x, p11[3].x};
    f4 V111 = {p11[0].y, p11[1].y, p11[2].y, p11[3].y};

    f4 X00 = V000 + fw * (V001 - V000);
    f4 X01 = V010 + fw * (V011 - V010);
    f4 X10 = V100 + fw * (V101 - V100);
    f4 X11 = V110 + fw * (V111 - V110);
    f4 Y0  = X00 + fh * (X01 - X00);
    f4 Y1  = X10 + fh * (X11 - X10);
    f4 O   = Y0  + fd * (Y1  - Y0);

    // Streaming output store (written once): non-temporal.
    __builtin_nontemporal_store(O, (f4*)(out + (size_t)b * cDHW + r));
}

extern "C" void kernel_launch(void* const* d_in, const int* in_sizes, int n_in,
                              void* d_out, int out_size, void* d_ws, size_t ws_size,
                              hipStream_t stream) {
    const float* src  = (const float*)d_in[0];
    const float* flow = (const float*)d_in[1];
    float*       out  = (float*)d_out;

    dim3 block(8, 8, 4);                 // 256 threads = 8 wave32
    dim3 grid(Wc / 32, Hc / 8, (Dc / 4) * Bc);   // (7, 24, 80) -> exact cover

    spatial_transformer_warp<<<grid, block, 0, stream>>>(src, flow, out);
}